// AnomalyAttention_new_51994874085550
// MI455X (gfx1250) — compile-verified
//
#include <hip/hip_runtime.h>

// Fused causal attention (AnomalyAttention forward), fp32 end-to-end using
// CDNA5 V_WMMA_F32_16X16X4_F32 (full-precision matrix pipe, wave32).
//
// B=2, C=4, H=8 -> 64 heads; L=1024, E=64. One wave per (head, 16-row tile),
// flash-attention online softmax in the exp2 domain, processing 32 columns
// (two 16x16 score tiles) per iteration to halve per-iteration softmax /
// rescale overhead per column.
//   - row-max:  DPP16 butterfly in the VALU pipe (no LDS permutes)
//   - row-sum:  WMMA against an all-ones B matrix (reuses P fragments)
//   - P C/D->A layout transpose through 2.2 KB of LDS
//   - global_prefetch of the next K/V tile pair

typedef __attribute__((ext_vector_type(2))) float v2f;
typedef __attribute__((ext_vector_type(8))) float v8f;

#define L_SEQ 1024
#define E_DIM 64
#define NHEAD 64                 // B*C*H
#define SCALE_LOG2E 0.18033688f  // (1/sqrt(64)) * log2(e)
#define NEG_BIG (-1e30f)
#define PT_STRIDE 34             // even -> 8B-aligned float2 reads, bank-spread

// max-reduce across the 16 lanes of a DPP row (our half-wave == DPP row)
__device__ __forceinline__ float rowmax16(float x) {
#define DPP_STEP(CTRL)                                                        \
    {                                                                         \
        int t_ = __builtin_amdgcn_update_dpp(                                 \
            0, __builtin_bit_cast(int, x), (CTRL), 0xF, 0xF, true);           \
        x = fmaxf(x, __builtin_bit_cast(float, t_));                          \
    }
    DPP_STEP(0xB1)   // quad_perm [1,0,3,2]  : xor 1
    DPP_STEP(0x4E)   // quad_perm [2,3,0,1]  : xor 2
    DPP_STEP(0x141)  // ROW_HALF_MIRROR      : combine quads within 8
    DPP_STEP(0x140)  // ROW_MIRROR           : combine 8-groups within 16
#undef DPP_STEP
    return x;
}

__global__ __launch_bounds__(32)
void aattn_fwd_f32wmma(const float* __restrict__ Q,
                       const float* __restrict__ K,
                       const float* __restrict__ V,
                       float* __restrict__ Out) {
    const int lane = threadIdx.x;      // 0..31
    const int half = lane >> 4;        // DPP row / K-pair select
    const int l16  = lane & 15;
    const int itile = blockIdx.x;      // row tile, 0..63
    const int head  = blockIdx.y;      // 0..63

    const size_t hb = (size_t)head * L_SEQ * E_DIM;
    const float* q  = Q + hb + (size_t)itile * 16 * E_DIM;
    const float* kp = K + hb;
    const float* vp = V + hb;
    float* o = Out + hb + (size_t)itile * 16 * E_DIM;

    // LDS bounce buffer: 16 rows x 32 cols (stride 34) for the P transpose.
    __shared__ float pt[16 * PT_STRIDE];

    // ---- Q tile as 16 A-fragments (16x4 f32), scaled by log2(e)/sqrt(E) ----
    v2f qa[16];
#pragma unroll
    for (int e = 0; e < 16; ++e) {
        const float* p = q + l16 * E_DIM + 4 * e + 2 * half;
        v2f t; t.x = p[0] * SCALE_LOG2E; t.y = p[1] * SCALE_LOG2E;
        qa[e] = t;
    }

    // ---- state: O (4 C/D tiles), l (C/D tile, cols replicated), m per row ----
    v8f acc[4] = {};
    v8f lacc = {};
    float mrow[8];
#pragma unroll
    for (int r = 0; r < 8; ++r) mrow[r] = NEG_BIG;

    const v2f ones = {1.0f, 1.0f};     // all-ones B fragment (layout-proof)

    // ---- shared tail of every step: transpose, l-WMMA, PV-WMMA ----
    // (s tiles already exponentiated; NT = number of 16-col tiles this step)
    auto softmax_core = [&](v8f* s, int NT, const float* vt) {
        // transpose P tiles through LDS: C/D layout -> A layout
        __syncthreads();   // single-wave WG: just orders DS ops
        for (int t = 0; t < NT; ++t)
#pragma unroll
            for (int r = 0; r < 8; ++r)
                pt[(r + 8 * half) * PT_STRIDE + 16 * t + l16] = s[t][r];
        __syncthreads();
        for (int t = 0; t < NT; ++t) {
            v2f pa[4];
#pragma unroll
            for (int kk = 0; kk < 4; ++kk) {
                const float* p = &pt[l16 * PT_STRIDE + 16 * t + 4 * kk + 2 * half];
                v2f w; w.x = p[0]; w.y = p[1];
                pa[kk] = w;
            }
            // l += P . 1 : row sums via WMMA (every column == row sum)
#pragma unroll
            for (int kk = 0; kk < 4; ++kk)
                lacc = __builtin_amdgcn_wmma_f32_16x16x4_f32(
                           false, pa[kk], false, ones, (short)0, lacc, false, false);
            // O += P . V
#pragma unroll
            for (int n = 0; n < 4; ++n) {
#pragma unroll
                for (int kk = 0; kk < 4; ++kk) {
                    const float* pv = vt + (16 * t + 4 * kk + 2 * half) * E_DIM
                                         + 16 * n + l16;
                    v2f b; b.x = pv[0]; b.y = pv[E_DIM];
                    acc[n] = __builtin_amdgcn_wmma_f32_16x16x4_f32(
                                 false, pa[kk], false, b, (short)0, acc[n],
                                 false, false);
                }
            }
        }
    };

    // ================= main loop: pairs of unmasked tiles =================
    int j = 0;
    for (; j + 2 <= itile; j += 2) {
        const float* kt = kp + (size_t)j * 16 * E_DIM;
        const float* vt = vp + (size_t)j * 16 * E_DIM;

        // prefetch the next 8KB K/V pair (one 128B line per lane x2)
        if (j + 4 <= itile) {
            __builtin_prefetch(kt + 32 * E_DIM + lane * 32, 0, 3);
            __builtin_prefetch(kt + 32 * E_DIM + 1024 + lane * 32, 0, 3);
            __builtin_prefetch(vt + 32 * E_DIM + lane * 32, 0, 3);
            __builtin_prefetch(vt + 32 * E_DIM + 1024 + lane * 32, 0, 3);
        }

        // ---- S0,S1 = Q . K^T for both tiles (32 WMMAs over E) ----
        v8f s[2] = {};
#pragma unroll
        for (int e = 0; e < 16; ++e) {
            const float* p0 = kt + l16 * E_DIM + 4 * e + 2 * half;
            v2f b0; b0.x = p0[0]; b0.y = p0[1];
            s[0] = __builtin_amdgcn_wmma_f32_16x16x4_f32(
                       false, qa[e], false, b0, (short)0, s[0], false, false);
            const float* p1 = p0 + 16 * E_DIM;
            v2f b1; b1.x = p1[0]; b1.y = p1[1];
            s[1] = __builtin_amdgcn_wmma_f32_16x16x4_f32(
                       false, qa[e], false, b1, (short)0, s[1], false, false);
        }

        // ---- online softmax over 32 columns at once ----
#pragma unroll
        for (int r = 0; r < 8; ++r) {
            const float rm    = rowmax16(fmaxf(s[0][r], s[1][r]));
            const float mnew  = fmaxf(mrow[r], rm);
            const float alpha = __builtin_amdgcn_exp2f(mrow[r] - mnew);
            mrow[r]  = mnew;
            s[0][r]  = __builtin_amdgcn_exp2f(s[0][r] - mnew);
            s[1][r]  = __builtin_amdgcn_exp2f(s[1][r] - mnew);
            lacc[r] *= alpha;
#pragma unroll
            for (int n = 0; n < 4; ++n) acc[n][r] *= alpha;
        }

        softmax_core(s, 2, vt);
    }

    // ============ leftover single unmasked tile (itile odd) ============
    // ============ then the masked diagonal tile (always) ===============
#pragma unroll 1
    for (; j <= itile; ++j) {
        const bool diag = (j == itile);
        const float* kt = kp + (size_t)j * 16 * E_DIM;
        const float* vt = vp + (size_t)j * 16 * E_DIM;

        v8f s[1] = {};
#pragma unroll
        for (int e = 0; e < 16; ++e) {
            const float* p = kt + l16 * E_DIM + 4 * e + 2 * half;
            v2f b; b.x = p[0]; b.y = p[1];
            s[0] = __builtin_amdgcn_wmma_f32_16x16x4_f32(
                       false, qa[e], false, b, (short)0, s[0], false, false);
        }

        if (diag) {   // causal mask: col > row within the diagonal tile
#pragma unroll
            for (int r = 0; r < 8; ++r) {
                const int row = r + 8 * half;
                if (l16 > row) s[0][r] = NEG_BIG;
            }
        }

#pragma unroll
        for (int r = 0; r < 8; ++r) {
            const float rm    = rowmax16(s[0][r]);
            const float mnew  = fmaxf(mrow[r], rm);
            const float alpha = __builtin_amdgcn_exp2f(mrow[r] - mnew);
            mrow[r]  = mnew;
            s[0][r]  = __builtin_amdgcn_exp2f(s[0][r] - mnew);
            lacc[r] *= alpha;
#pragma unroll
            for (int n = 0; n < 4; ++n) acc[n][r] *= alpha;
        }

        softmax_core(s, 1, vt);
    }

    // ---- normalize by row sums and store ----
#pragma unroll
    for (int r = 0; r < 8; ++r) {
        const float inv = 1.0f / lacc[r];
        const int row = r + 8 * half;
#pragma unroll
        for (int n = 0; n < 4; ++n)
            o[row * E_DIM + 16 * n + l16] = acc[n][r] * inv;
    }
}

extern "C" void kernel_launch(void* const* d_in, const int* in_sizes, int n_in,
                              void* d_out, int out_size, void* d_ws, size_t ws_size,
                              hipStream_t stream) {
    // inputs: queries, keys, values (f32), attn_mask (bool, statically causal -> ignored)
    const float* Q = (const float*)d_in[0];
    const float* K = (const float*)d_in[1];
    const float* V = (const float*)d_in[2];
    float* O = (float*)d_out;
    (void)in_sizes; (void)n_in; (void)out_size; (void)d_ws; (void)ws_size;

    dim3 grid(L_SEQ / 16, NHEAD);   // 64 row-tiles x 64 heads = 4096 waves
    dim3 block(32);                 // one wave32 per workgroup
    aattn_fwd_f32wmma<<<grid, block, 0, stream>>>(Q, K, V, O);
}